// BackgroundNoiseLayer_33380485825261
// MI455X (gfx1250) — compile-verified
//
#include <hip/hip_runtime.h>

// ---------------- problem constants (match reference) ----------------
#define N_NEURONS 40000
#define N_BKG     100      // K of the GEMM
#define NNZ       200000
#define NSB       5        // n_syn_basis
#define SEQ       600      // T of the GEMM
#define TPAD      608      // T padded to 19 * 32
#define NC        (N_NEURONS * NSB)   // 200000 GEMM columns
#define W_ELEMS   (N_BKG * NC)        // 20,000,000 floats = 80 MB

typedef __attribute__((ext_vector_type(2))) float v2f;
typedef __attribute__((ext_vector_type(8))) float v8f;

// ---------------- phase 1: zero the dense W in workspace ----------------
__global__ __launch_bounds__(256) void zero_W_kernel(float4* __restrict__ W4) {
    int i = blockIdx.x * 256 + threadIdx.x;
    if (i < W_ELEMS / 4) W4[i] = make_float4(0.f, 0.f, 0.f, 0.f);
}

// ---------------- phase 2: spikes (padded to TPAD rows) ----------------
__global__ __launch_bounds__(256) void spikes_kernel(const float* __restrict__ rob,
                                                     float* __restrict__ S) {
    int i = blockIdx.x * 256 + threadIdx.x;
    if (i >= TPAD * N_BKG) return;
    int t = i / N_BKG;
    float v = 0.f;
    if (t < SEQ) v = (rob[i] < 0.25f) ? 1.f : 0.f;   // BKG_RATE * 0.001
    S[i] = v;
}

// ---------------- phase 3: sparse scatter-add into W[k][n*5+r] ----------------
__global__ __launch_bounds__(256) void scatter_kernel(const float* __restrict__ weights,
                                                      const float* __restrict__ synw,
                                                      const int*   __restrict__ post_idx,
                                                      const int*   __restrict__ pre_idx,
                                                      const int*   __restrict__ syn_ids,
                                                      float*       __restrict__ W) {
    int e = blockIdx.x * 256 + threadIdx.x;
    if (e >= NNZ) return;
    float we = weights[e];
    int p = post_idx[e];          // [0, 40000)
    int q = pre_idx[e];           // [0, 100)  -> GEMM row (k)
    int s = syn_ids[e];           // [0, 10)
    float* base = W + (size_t)q * NC + (size_t)p * NSB;
    const float* f = synw + s * NSB;
#pragma unroll
    for (int r = 0; r < NSB; ++r) {
        unsafeAtomicAdd(base + r, we * f[r]);   // global_atomic_add_f32
    }
}

// ---------------- phase 4: WMMA fp32 GEMM  out(T x NC) = S(T x K) * W(K x NC) ----
// One wave computes a 32(t) x 16(c) tile: two 16x16 accumulators sharing each
// B fragment. V_WMMA_F32_16X16X4_F32, 25 K-steps of 4.
__global__ __launch_bounds__(256) void gemm_wmma_kernel(const float* __restrict__ S,
                                                        const float* __restrict__ W,
                                                        float* __restrict__ out) {
    const int NCT = NC / 16;          // 12500 column tiles
    const int NTG = TPAD / 32;        // 19 t-groups
    int wave = (blockIdx.x * 256 + threadIdx.x) >> 5;  // wave-uniform
    if (wave >= NTG * NCT) return;                     // uniform exit: EXEC stays all-1s
    int lane = threadIdx.x & 31;
    int lo = lane & 15;               // M / N index within half-wave
    int hi = lane >> 4;               // selects K pair / M+8 half

    int tg = wave / NCT;
    int ct = wave - tg * NCT;
    int cBase = ct * 16;
    int tb0 = tg * 32;                // first 16-row t-tile
    int tb1 = tb0 + 16;               // second 16-row t-tile

    // B column pointer: lane -> column cBase+lo, rows strided by NC
    const float* Wc = W + cBase + lo;
    // A row pointers: lane -> t-row tb+lo
    const float* S0 = S + (size_t)(tb0 + lo) * N_BKG;
    const float* S1 = S + (size_t)(tb1 + lo) * N_BKG;

    v8f acc0 = {0.f, 0.f, 0.f, 0.f, 0.f, 0.f, 0.f, 0.f};
    v8f acc1 = {0.f, 0.f, 0.f, 0.f, 0.f, 0.f, 0.f, 0.f};

    for (int kk = 0; kk < N_BKG; kk += 4) {
        int kA = kk + 2 * hi;         // ISA layout: VGPR0 holds K = 2*(lane/16), VGPR1 = +1
        v2f b, a0, a1;
        b.x  = Wc[(size_t)kA * NC];
        b.y  = Wc[(size_t)(kA + 1) * NC];
        a0.x = S0[kA];
        a0.y = S0[kA + 1];
        a1.x = S1[kA];
        a1.y = S1[kA + 1];
        // 8 args: (neg_a, A, neg_b, B, c_mod, C, reuse_a, reuse_b)
        acc0 = __builtin_amdgcn_wmma_f32_16x16x4_f32(false, a0, false, b,
                                                     (short)0, acc0, false, false);
        acc1 = __builtin_amdgcn_wmma_f32_16x16x4_f32(false, a1, false, b,
                                                     (short)0, acc1, false, false);
    }

    // D layout: VGPR v, lanes 0-15 -> M=v, lanes 16-31 -> M=v+8; N = lane%16
    int c = cBase + lo;
#pragma unroll
    for (int v = 0; v < 8; ++v) {
        int t0 = tb0 + v + 8 * hi;    // tb0 <= 576 -> t0 <= 591 < 600: always valid
        out[(size_t)t0 * NC + c] = acc0[v];
        int t1 = tb1 + v + 8 * hi;    // up to 607: mask padded rows (after all WMMAs)
        if (t1 < SEQ) out[(size_t)t1 * NC + c] = acc1[v];
    }
}

// ---------------- launcher ----------------
extern "C" void kernel_launch(void* const* d_in, const int* in_sizes, int n_in,
                              void* d_out, int out_size, void* d_ws, size_t ws_size,
                              hipStream_t stream) {
    const float* weights  = (const float*)d_in[0];   // (200000,)
    const float* synw     = (const float*)d_in[1];   // (10, 5)
    const float* rob      = (const float*)d_in[2];   // (1, 600, 100)
    const int*   post_idx = (const int*)d_in[3];     // (200000,)
    const int*   pre_idx  = (const int*)d_in[4];     // (200000,)
    const int*   syn_ids  = (const int*)d_in[5];     // (200000,)

    float* W = (float*)d_ws;                 // 20,000,000 floats (80 MB), K-major (100 x 200000)
    float* S = W + W_ELEMS;                  // 608 x 100 padded spike matrix
    float* out = (float*)d_out;              // 600 x 200000

    zero_W_kernel<<<(W_ELEMS / 4 + 255) / 256, 256, 0, stream>>>((float4*)W);
    spikes_kernel<<<(TPAD * N_BKG + 255) / 256, 256, 0, stream>>>(rob, S);
    scatter_kernel<<<(NNZ + 255) / 256, 256, 0, stream>>>(weights, synw, post_idx,
                                                          pre_idx, syn_ids, W);

    const int totalWaves = (TPAD / 32) * (NC / 16);  // 19 * 12500 = 237500
    gemm_wmma_kernel<<<(totalWaves + 7) / 8, 256, 0, stream>>>(S, W, out);
}